// MSE_OHEM_Loss_1022202217305
// MI455X (gfx1250) — compile-verified
//
#include <hip/hip_runtime.h>
#include <hip/hip_bf16.h>
#include <stdint.h>

// ---------------------------------------------------------------------------
// OHEM MSE loss, 3-level radix-select top-k, CDNA5 (gfx1250), wave32.
// 64 problems (32 batch x 2 channels), N = 640*640 elements each.
// Passes: [1] stats + L1 hist (async-to-LDS staged)  [2] L2 hist  [3] L3 hist
//         [4] sum of losses strictly above exact threshold  [5] combine.
// ---------------------------------------------------------------------------

#define BATCH   32
#define OUTW    640
#define N_PIX   (OUTW * OUTW)        // 409600
#define TGW     320
#define NPROB   (BATCH * 2)          // 64
#define NB      16                   // blocks per problem in streaming passes
#define THREADS 256
#define CHUNK   (NB * THREADS * 4)   // 16384 elements per sweep of a problem
#define ITERS   (N_PIX / CHUNK)      // 25
#define HISTB   4096                 // per-problem histogram slice (max level)

// ctl word indices (16 u32 per problem)
#define C_NUMPOS 0
#define C_K      1
#define C_FULL   2
#define C_BIN1   3
#define C_NEED1  4
#define C_PFX20  5
#define C_NEED2  6
#define C_THR    7
#define C_REM    8

typedef __attribute__((ext_vector_type(2))) float v2f;
typedef __attribute__((ext_vector_type(8))) float v8f;

// ---------------- WMMA f32 wave reduction (exact f32 adds) -----------------
// A 16x4: lanes0-15 vgpr0 = K0, lanes16-31 vgpr0 = K2 (vgpr1 = K1/K3 = 0).
// B 4x16 = ones  =>  D[m][n] = v(m) + v(m+16), broadcast over n.
// Sum D's 8 VGPRs per lane, combine low/high half-wave with one shfl_xor(16).
__device__ __forceinline__ float wave_reduce_wmma(float v) {
  v2f a; a.x = v;    a.y = 0.0f;
  v2f b; b.x = 1.0f; b.y = 1.0f;
  v8f c = {};
  c = __builtin_amdgcn_wmma_f32_16x16x4_f32(false, a, false, b,
                                            (short)0, c, false, false);
  float s = c[0] + c[1] + c[2] + c[3] + c[4] + c[5] + c[6] + c[7];
  s += __shfl_xor(s, 16, 32);
  return s;  // full wave sum in every lane
}

__device__ __forceinline__ float block_reduce_wmma(float v, float* lred) {
  float w = wave_reduce_wmma(v);
  int wid  = threadIdx.x >> 5;
  int lane = threadIdx.x & 31;
  if (lane == 0) lred[wid] = w;
  __syncthreads();
  float r = 0.0f;
  if (threadIdx.x == 0) {
#pragma unroll
    for (int i = 0; i < THREADS / 32; ++i) r += lred[i];
  }
  __syncthreads();
  return r;  // valid in thread 0
}

// ---------------- bilinear 2x upsample, half-pixel centers -----------------
// even out pixel 2j: 0.25*t[j-1] + 0.75*t[j];  odd 2j+1: 0.75*t[j] + 0.25*t[j+1]
__device__ __forceinline__ void lin_coef(int o, int& i0, int& i1,
                                         float& w0, float& w1) {
  int j = o >> 1;
  if (o & 1) { i0 = j; i1 = (j + 1 < TGW) ? j + 1 : TGW - 1; w0 = 0.75f; w1 = 0.25f; }
  else       { i0 = (j > 0) ? j - 1 : 0; i1 = j;             w0 = 0.25f; w1 = 0.75f; }
}

__device__ __forceinline__ float4 upsample4(const float* __restrict__ t,
                                            int y, int x4) {
  int y0, y1; float wy0, wy1;
  lin_coef(y, y0, y1, wy0, wy1);
  const float* r0 = t + y0 * TGW;
  const float* r1 = t + y1 * TGW;
  int m  = x4 >> 2;  // x4 = 4m; pixels 4m..4m+3 need target cols 2m-1..2m+2
  int c0 = (2 * m - 1 > 0) ? 2 * m - 1 : 0;
  int c1 = 2 * m;
  int c2 = 2 * m + 1;
  int c3 = (2 * m + 2 < TGW) ? 2 * m + 2 : TGW - 1;
  float v0 = wy0 * r0[c0] + wy1 * r1[c0];
  float v1 = wy0 * r0[c1] + wy1 * r1[c1];
  float v2 = wy0 * r0[c2] + wy1 * r1[c2];
  float v3 = wy0 * r0[c3] + wy1 * r1[c3];
  return make_float4(0.25f * v0 + 0.75f * v1,
                     0.75f * v1 + 0.25f * v2,
                     0.25f * v1 + 0.75f * v2,
                     0.75f * v2 + 0.25f * v3);
}

__device__ __forceinline__ const float* problem_pred(const float* x, int p) {
  return x + (size_t)p * N_PIX;  // p = b*2 + ch, x is [B][2][640][640]
}
__device__ __forceinline__ const float* problem_tgt(const float* ct,
                                                    const float* at, int p) {
  int b = p >> 1, ch = p & 1;
  return (ch == 0 ? ct : at) + (size_t)b * (TGW * TGW);
}

// ---------------- pass 1: stats + level-1 histogram (async staged) ---------
__global__ void __launch_bounds__(THREADS)
pass1_kernel(const float* __restrict__ x, const float* __restrict__ ctgt,
             const float* __restrict__ atgt, unsigned* __restrict__ hist,
             unsigned* __restrict__ ctl, float* __restrict__ partA,
             float* __restrict__ partP) {
  __shared__ unsigned lh[1024];
  __shared__ float lred[THREADS / 32];
  __shared__ unsigned cPosSh;
  __shared__ __align__(16) float stage[2][THREADS * 4];

  const int p = blockIdx.x / NB, sub = blockIdx.x % NB;
  const int tid = threadIdx.x;
  const float* pred = problem_pred(x, p);
  const float* tgt  = problem_tgt(ctgt, atgt, p);

  for (int i = tid; i < 1024; i += THREADS) lh[i] = 0;
  if (tid == 0) cPosSh = 0;
  __syncthreads();

  const unsigned stageBase = (unsigned)(uintptr_t)(&stage[0][0]);
  float sAll = 0.0f, sPos = 0.0f;
  unsigned cPos = 0;

  // issue async copy of this thread's 16B chunk for iteration `it` into buf
  auto issue = [&](int it, int buf) {
    int idx = it * CHUNK + sub * (THREADS * 4) + tid * 4;
    unsigned ldsAddr = stageBase + (unsigned)((buf * (THREADS * 4) + tid * 4) * 4);
    int byteOff = idx * 4;
    asm volatile("global_load_async_to_lds_b128 %0, %1, %2 offset:0"
                 :: "v"(ldsAddr), "v"(byteOff), "s"(pred) : "memory");
  };

  issue(0, 0);
  for (int it = 0; it < ITERS; ++it) {
    const int buf = it & 1;
    if (it + 1 < ITERS) {
      issue(it + 1, buf ^ 1);
      asm volatile("s_wait_asynccnt 1" ::: "memory");
    } else {
      asm volatile("s_wait_asynccnt 0" ::: "memory");
    }
    float4 pr = *reinterpret_cast<const float4*>(&stage[buf][tid * 4]);
    int idx = it * CHUNK + sub * (THREADS * 4) + tid * 4;
    int y = idx / OUTW, x4 = idx - y * OUTW;
    float4 tv = upsample4(tgt, y, x4);
#pragma unroll
    for (int c = 0; c < 4; ++c) {
      float prv = (&pr.x)[c], tvv = (&tv.x)[c];
      float d = prv - tvv, l = d * d;
      sAll += l;
      if (tvv > 0.0f) { cPos++; sPos += l; }
      else atomicAdd(&lh[__float_as_uint(l) >> 22], 1u);
    }
  }
  __syncthreads();
  for (int i = tid; i < 1024; i += THREADS)
    if (lh[i]) atomicAdd(&hist[(size_t)p * HISTB + i], lh[i]);
  atomicAdd(&cPosSh, cPos);
  __syncthreads();
  if (tid == 0) atomicAdd(&ctl[p * 16 + C_NUMPOS], cPosSh);

  float bA = block_reduce_wmma(sAll, lred);
  if (tid == 0) partA[p * NB + sub] = bA;
  float bP = block_reduce_wmma(sPos, lred);
  if (tid == 0) partP[p * NB + sub] = bP;
}

// ---------------- scan: pick bin holding the k-th largest, then zero hist --
__global__ void scan_kernel(unsigned* __restrict__ hist,
                            unsigned* __restrict__ ctl, int level) {
  const int p = blockIdx.x;
  unsigned* h = hist + (size_t)p * HISTB;
  unsigned* c = ctl + p * 16;
  const int bins = (level == 2) ? 4096 : 1024;
  if (threadIdx.x == 0) {
    if (level == 0) {
      int np = (int)c[C_NUMPOS];
      int k = (int)((float)np * 3.0f);      // trunc toward zero, like astype
      int lim = N_PIX - np;
      if (k > lim) k = lim;
      c[C_K] = (unsigned)k;
      c[C_FULL] = (k < 10) ? 1u : 0u;
    }
    if (c[C_FULL] == 0u) {
      int need = (level == 0) ? (int)c[C_K]
               : (level == 1) ? (int)c[C_NEED1] : (int)c[C_NEED2];
      int cum = 0, selb = 0;
      for (int bI = bins - 1; bI >= 0; --bI) {
        int cb = (int)h[bI];
        if (cum + cb >= need) { selb = bI; need -= cum; break; }
        cum += cb;
      }
      if (level == 0)      { c[C_BIN1] = (unsigned)selb; c[C_NEED1] = (unsigned)need; }
      else if (level == 1) { c[C_PFX20] = (c[C_BIN1] << 10) | (unsigned)selb;
                             c[C_NEED2] = (unsigned)need; }
      else                 { c[C_THR] = (c[C_PFX20] << 12) | (unsigned)selb;
                             c[C_REM] = (unsigned)need; }
    }
  }
  __syncthreads();
  for (int i = threadIdx.x; i < bins; i += blockDim.x) h[i] = 0;
}

// ---------------- passes 2/3: refine histogram within the selected bin -----
__global__ void __launch_bounds__(THREADS)
passmid_kernel(const float* __restrict__ x, const float* __restrict__ ctgt,
               const float* __restrict__ atgt, unsigned* __restrict__ hist,
               const unsigned* __restrict__ ctl, int level) {
  const int p = blockIdx.x / NB, sub = blockIdx.x % NB;
  const unsigned* c = ctl + p * 16;
  if (c[C_FULL]) return;  // uniform per block
  __shared__ unsigned lh[HISTB];
  const int bins = (level == 1) ? 1024 : 4096;
  const unsigned sel = (level == 1) ? c[C_BIN1] : c[C_PFX20];
  for (int i = threadIdx.x; i < bins; i += THREADS) lh[i] = 0;
  __syncthreads();
  const float* pred = problem_pred(x, p);
  const float* tgt  = problem_tgt(ctgt, atgt, p);
  for (int it = 0; it < ITERS; ++it) {
    int idx = it * CHUNK + sub * (THREADS * 4) + threadIdx.x * 4;
    float4 pr = *reinterpret_cast<const float4*>(pred + idx);
    int y = idx / OUTW, x4 = idx - y * OUTW;
    float4 tv = upsample4(tgt, y, x4);
#pragma unroll
    for (int cc = 0; cc < 4; ++cc) {
      float prv = (&pr.x)[cc], tvv = (&tv.x)[cc];
      float d = prv - tvv, l = d * d;
      if (tvv <= 0.0f) {
        unsigned bv = __float_as_uint(l);
        if (level == 1) { if ((bv >> 22) == sel) atomicAdd(&lh[(bv >> 12) & 0x3FFu], 1u); }
        else            { if ((bv >> 12) == sel) atomicAdd(&lh[bv & 0xFFFu], 1u); }
      }
    }
  }
  __syncthreads();
  for (int i = threadIdx.x; i < bins; i += THREADS)
    if (lh[i]) atomicAdd(&hist[(size_t)p * HISTB + i], lh[i]);
}

// ---------------- pass 4: sum of negative losses strictly above threshold --
__global__ void __launch_bounds__(THREADS)
pass4_kernel(const float* __restrict__ x, const float* __restrict__ ctgt,
             const float* __restrict__ atgt, const unsigned* __restrict__ ctl,
             float* __restrict__ partG) {
  const int p = blockIdx.x / NB, sub = blockIdx.x % NB;
  const unsigned* c = ctl + p * 16;
  if (c[C_FULL]) return;  // partG slot stays 0 from the zero pass
  __shared__ float lred[THREADS / 32];
  const unsigned T = c[C_THR];
  const float* pred = problem_pred(x, p);
  const float* tgt  = problem_tgt(ctgt, atgt, p);
  float sG = 0.0f;
  for (int it = 0; it < ITERS; ++it) {
    int idx = it * CHUNK + sub * (THREADS * 4) + threadIdx.x * 4;
    float4 pr = *reinterpret_cast<const float4*>(pred + idx);
    int y = idx / OUTW, x4 = idx - y * OUTW;
    float4 tv = upsample4(tgt, y, x4);
#pragma unroll
    for (int cc = 0; cc < 4; ++cc) {
      float prv = (&pr.x)[cc], tvv = (&tv.x)[cc];
      float d = prv - tvv, l = d * d;
      if (tvv <= 0.0f && __float_as_uint(l) > T) sG += l;
    }
  }
  float bG = block_reduce_wmma(sG, lred);
  if (threadIdx.x == 0) partG[p * NB + sub] = bG;
}

// ---------------- finalize: per-problem loss, mean over batch --------------
__global__ void final_kernel(const unsigned* __restrict__ ctl,
                             const float* __restrict__ partA,
                             const float* __restrict__ partP,
                             const float* __restrict__ partG,
                             float* __restrict__ out) {
  __shared__ float acc[NPROB];
  int t = threadIdx.x;  // 64 threads, one per problem
  const unsigned* c = ctl + t * 16;
  float sA = 0.0f, sP = 0.0f, sG = 0.0f;
  for (int j = 0; j < NB; ++j) {
    sA += partA[t * NB + j];
    sP += partP[t * NB + j];
    sG += partG[t * NB + j];
  }
  float lossP;
  if (c[C_FULL]) {
    lossP = sA / (float)N_PIX;
  } else {
    int np = (int)c[C_NUMPOS];
    int k  = (int)c[C_K];
    float posMean = sP / (float)(np > 1 ? np : 1);
    float topSum  = sG + (float)c[C_REM] * __uint_as_float(c[C_THR]);
    lossP = posMean + topSum / (float)(k > 1 ? k : 1);
  }
  acc[t] = lossP;
  __syncthreads();
  if (t == 0) {
    float s = 0.0f;
    for (int i = 0; i < NPROB; ++i) s += acc[i];
    out[0] = s / (float)BATCH;
  }
}

__global__ void zero_kernel(unsigned* __restrict__ w, int n) {
  int i = blockIdx.x * blockDim.x + threadIdx.x;
  if (i < n) w[i] = 0;
}

// ---------------------------------------------------------------------------
extern "C" void kernel_launch(void* const* d_in, const int* in_sizes, int n_in,
                              void* d_out, int out_size, void* d_ws, size_t ws_size,
                              hipStream_t stream) {
  (void)in_sizes; (void)n_in; (void)out_size; (void)ws_size;
  const float* x  = (const float*)d_in[0];
  const float* ct = (const float*)d_in[1];
  const float* at = (const float*)d_in[2];
  float* out = (float*)d_out;

  // workspace layout (~1.05 MB): hist | ctl | partA | partP | partG
  unsigned* hist = (unsigned*)d_ws;
  unsigned* ctl  = hist + (size_t)NPROB * HISTB;
  float* partA = (float*)(ctl + NPROB * 16);
  float* partP = partA + NPROB * NB;
  float* partG = partP + NPROB * NB;
  const int wsWords = NPROB * HISTB + NPROB * 16 + 3 * NPROB * NB;

  zero_kernel<<<(wsWords + 255) / 256, 256, 0, stream>>>((unsigned*)d_ws, wsWords);
  pass1_kernel<<<NPROB * NB, THREADS, 0, stream>>>(x, ct, at, hist, ctl, partA, partP);
  scan_kernel<<<NPROB, 256, 0, stream>>>(hist, ctl, 0);
  passmid_kernel<<<NPROB * NB, THREADS, 0, stream>>>(x, ct, at, hist, ctl, 1);
  scan_kernel<<<NPROB, 256, 0, stream>>>(hist, ctl, 1);
  passmid_kernel<<<NPROB * NB, THREADS, 0, stream>>>(x, ct, at, hist, ctl, 2);
  scan_kernel<<<NPROB, 256, 0, stream>>>(hist, ctl, 2);
  pass4_kernel<<<NPROB * NB, THREADS, 0, stream>>>(x, ct, at, ctl, partG);
  final_kernel<<<1, NPROB, 0, stream>>>(ctl, partA, partP, partG, out);
}